// Cache_55800215110244
// MI455X (gfx1250) — compile-verified
//
#include <hip/hip_runtime.h>
#include <stdint.h>

// Problem constants from the reference: B=8, S=4096, CHUNK=128, D=1024, fp32.
#define B_      8
#define S_      4096
#define CHUNK_  128
#define D_      1024
#define ROW_BYTES (D_ * 4)     // 4096 bytes per (b, s) row
#define THREADS   256          // 8 wave32s; 256 lanes * 16B = one full row
#define BATCH     8            // rows in flight per block (8 LDS buffers, 32 KiB)

// Pure data-movement kernel: out = cache, except rows s in [idx, idx+CHUNK)
// which come from value. Implemented as a global->LDS->global async DMA
// pipeline (CDNA5 ASYNCcnt path): payload never touches VGPRs, each wave keeps
// BATCH 512B requests in flight per phase, and the only synchronization is
// s_wait_asynccnt (no barriers: every thread owns a private 16B LDS slot).
__global__ __launch_bounds__(THREADS) void cache_update_async(
    const float* __restrict__ cache,
    const float* __restrict__ value,
    const int*   __restrict__ index,   // low 32 bits of the int64 index
    float* __restrict__ out)
{
    __shared__ char smem[BATCH * ROW_BYTES];   // 32 KiB of 320 KiB WGP LDS

    const int tid = threadIdx.x;
    const int idx = index[0];                  // 0 <= idx < S-CHUNK

    // Workgroup-relative LDS byte offset of this thread's 16B slot in buffer 0.
    // (Low 32 bits of a flat pointer into LDS == WG-relative offset; the async
    //  instructions add the wave's LDS_BASE in hardware.)
    const uint32_t lds0 = (uint32_t)(uintptr_t)(&smem[0]) + (uint32_t)tid * 16u;

    const int rows_total = B_ * S_;            // 32768 rows

    for (int base = (int)blockIdx.x * BATCH; base < rows_total;
         base += (int)gridDim.x * BATCH) {

        // ---- Stage 1: issue BATCH async loads (global -> LDS, streaming) ---
#pragma unroll
        for (int j = 0; j < BATCH; ++j) {
            const int row = base + j;
            const int b  = row >> 12;          // / S_
            const int s  = row & (S_ - 1);     // % S_
            const uint32_t rel = (uint32_t)(s - idx);   // wave-uniform branch
            const float* src = (rel < (uint32_t)CHUNK_)
                ? (value + ((size_t)b * CHUNK_ + rel) * D_)
                : (cache + (size_t)row * D_);
            const unsigned long long ga =
                (unsigned long long)(uintptr_t)(src + (size_t)tid * 4);
            const uint32_t la = lds0 + (uint32_t)(j * ROW_BYTES);
            asm volatile("global_load_async_to_lds_b128 %0, %1, off th:TH_LOAD_NT"
                         :: "v"(la), "v"(ga) : "memory");
        }

        // All loads complete: loads retire in order and no stores are
        // outstanding here, so asynccnt==0 <=> all LDS buffers written.
        asm volatile("s_wait_asynccnt 0x0" ::: "memory");

        // ---- Stage 2: issue BATCH async stores (LDS -> global) ------------
#pragma unroll
        for (int j = 0; j < BATCH; ++j) {
            const int row = base + j;
            const unsigned long long ga = (unsigned long long)(uintptr_t)(
                out + (size_t)row * D_ + (size_t)tid * 4);
            const uint32_t la = lds0 + (uint32_t)(j * ROW_BYTES);
            asm volatile("global_store_async_from_lds_b128 %0, %1, off"
                         :: "v"(ga), "v"(la) : "memory");
        }

        // Drain stores before the next batch overwrites the LDS buffers (WAR).
        asm volatile("s_wait_asynccnt 0x0" ::: "memory");
    }
}

extern "C" void kernel_launch(void* const* d_in, const int* in_sizes, int n_in,
                              void* d_out, int out_size, void* d_ws, size_t ws_size,
                              hipStream_t stream) {
    const float* cache = (const float*)d_in[0];   // (8, 4096, 1024) f32
    const float* value = (const float*)d_in[1];   // (8, 128, 1024) f32
    const int*   index = (const int*)d_in[2];     // int64 scalar; low word valid
    float* out = (float*)d_out;                   // (8, 4096, 1024) f32

    (void)in_sizes; (void)n_in; (void)out_size; (void)d_ws; (void)ws_size;

    // 32768 rows / (BATCH=8) = 4096 batches; 2048 blocks -> 2 batches each.
    dim3 grid(2048), block(THREADS);
    hipLaunchKernelGGL(cache_update_async, grid, block, 0, stream,
                       cache, value, index, out);
}